// FastLinearMemory_71468255805733
// MI455X (gfx1250) — compile-verified
//
#include <hip/hip_runtime.h>
#include <math.h>

#define EPSF 1e-5f

typedef __attribute__((ext_vector_type(2))) float v2f;
typedef __attribute__((ext_vector_type(8))) float v8f;
typedef __attribute__((ext_vector_type(4))) unsigned int v4u;
typedef __attribute__((ext_vector_type(4))) int v4i;
typedef __attribute__((ext_vector_type(8))) int v8i;

__device__ __forceinline__ v8f v8f_zero() {
  v8f z;
#pragma unroll
  for (int i = 0; i < 8; ++i) z[i] = 0.0f;
  return z;
}

// D = A(16x4 f32) x B(4x16 f32) + C  -- exact f32 WMMA path on gfx1250
__device__ __forceinline__ v8f wmma_f32_k4(v2f a, v2f b, v8f c) {
  return __builtin_amdgcn_wmma_f32_16x16x4_f32(
      false, a, false, b, (short)0, c, false, false);
}

// LDS byte offset of a __shared__ pointer: generic LDS addresses carry the
// offset in the low 32 bits (aperture check only examines addr[63:32]).
__device__ __forceinline__ unsigned lds_off(const void* p) {
  return (unsigned)(uintptr_t)p;
}

// TDM: async-load one contiguous 4KB tile (1024 f32) global -> LDS.
// D# per ISA 8.3/8.4: count=1, type=2; data_size=4B; 1-D tile_dim0=1024.
// This toolchain exposes the 6-arg builtin (g0, g1, g2, g3, g4, cpol).
__device__ __forceinline__ void tdm_load_tile_4k(const float* gsrc, unsigned lds_byte) {
  unsigned long long ga = (unsigned long long)(uintptr_t)gsrc;
  v4u g0;
  g0[0] = 1u;                                              // count=1, user mode
  g0[1] = lds_byte;                                        // lds_addr
  g0[2] = (unsigned)(ga & 0xFFFFFFFFu);                    // global_addr lo
  g0[3] = (unsigned)((ga >> 32) & 0x1FFFFFFu) | (2u << 30);// global_addr hi | type=2
  v8i g1;
  g1[0] = (int)(2u << 16);          // data_size = 4 bytes; no multicast/pad/iter
  g1[1] = (int)(1024u << 16);       // tensor_dim0 = 1024 (bits 79:48, low half)
  g1[2] = (int)(1u << 16);          // tensor_dim0 hi=0 | tensor_dim1 = 1
  g1[3] = (int)(1024u << 16);       // tensor_dim1 hi=0 | tile_dim0 = 1024
  g1[4] = 0;                        // tile_dim1 = 0 (unused), tile_dim2 = 0
  g1[5] = 1024;                     // tensor_dim0_stride lo
  g1[6] = 0;
  g1[7] = 0;
  v4i gz4 = {0, 0, 0, 0};
  v8i gz8 = {0, 0, 0, 0, 0, 0, 0, 0};
  __builtin_amdgcn_tensor_load_to_lds(g0, g1, gz4, gz4, gz8, 0);
}

// ---------------------------------------------------------------------------
// Elementwise relu (qa = relu(addresses))
// ---------------------------------------------------------------------------
__global__ __launch_bounds__(256) void relu_kernel(const float* __restrict__ in,
                                                   float* __restrict__ out, int n) {
  int i = blockIdx.x * 256 + threadIdx.x;
  if (i < n) out[i] = fmaxf(in[i], 0.0f);
}

// ---------------------------------------------------------------------------
// C[M,N] = act(A[M,K] @ W[K,N] + bias[N]); dims multiples of 32/64.
// 256 thr = 8 waves; block tile 32x64; one 16x16 WMMA tile per wave.
// Fragments preloaded per K-chunk; two accumulators break the WMMA RAW chain.
// ---------------------------------------------------------------------------
__global__ __launch_bounds__(256) void gemm_bias_act(
    const float* __restrict__ A, const float* __restrict__ W,
    const float* __restrict__ bias, float* __restrict__ C,
    int M, int N, int K, int do_relu) {
  __shared__ float As[32][33];
  __shared__ float Bs[32][80];

  const int tid  = threadIdx.x;
  const int lane = tid & 31;
  const int wave = tid >> 5;
  const int row0 = blockIdx.y * 32;
  const int col0 = blockIdx.x * 64;
  const int wr = (wave >> 2) << 4;
  const int wc = (wave & 3) << 4;
  const int mn = lane & 15;
  const int kh = (lane >> 4) << 1;
  const int cm = (lane < 16) ? 0 : 8;

  v8f acc0 = v8f_zero(), acc1 = v8f_zero();

  for (int k0 = 0; k0 < K; k0 += 32) {
    {  // A tile 32x32
      int r  = tid >> 3;
      int kk = (tid & 7) << 2;
      const float* src = A + (size_t)(row0 + r) * K + k0 + kk;
      As[r][kk + 0] = src[0]; As[r][kk + 1] = src[1];
      As[r][kk + 2] = src[2]; As[r][kk + 3] = src[3];
    }
    {  // B tile 32x64
      int kk = tid >> 3;
      int nn = (tid & 7) << 3;
      const float* src = W + (size_t)(k0 + kk) * N + col0 + nn;
#pragma unroll
      for (int j = 0; j < 8; ++j) Bs[kk][nn + j] = src[j];
    }
    __syncthreads();

    v2f af[8], bf[8];
#pragma unroll
    for (int i = 0; i < 8; ++i) {   // batch all LDS fragment loads
      int kb = i << 2;
      af[i].x = As[wr + mn][kb + kh];
      af[i].y = As[wr + mn][kb + kh + 1];
      bf[i].x = Bs[kb + kh][wc + mn];
      bf[i].y = Bs[kb + kh + 1][wc + mn];
    }
#pragma unroll
    for (int i = 0; i < 8; i += 2) {  // two independent WMMA chains
      acc0 = wmma_f32_k4(af[i],     bf[i],     acc0);
      acc1 = wmma_f32_k4(af[i + 1], bf[i + 1], acc1);
    }
    __syncthreads();
  }

#pragma unroll
  for (int r = 0; r < 8; ++r) {
    int row = row0 + wr + cm + r;
    int col = col0 + wc + mn;
    float v = acc0[r] + acc1[r] + bias[col];
    if (do_relu) v = fmaxf(v, 0.0f);
    C[(size_t)row * N + col] = v;
  }
}

// ---------------------------------------------------------------------------
// Fused per-batch memory kernel: one block (8 waves) per batch b.
// memories tiles (the only big HBM stream) are double-buffered in LDS via the
// Tensor Data Mover, issued by wave 0 one iteration ahead (s_wait_tensorcnt).
// ---------------------------------------------------------------------------
__global__ __launch_bounds__(256) void fused_memory_kernel(
    const float* __restrict__ fk_all,     // (B, 512)  relu(x@Wk)
    const float* __restrict__ v_all,      // (B, 1024) x@Wv
    const float* __restrict__ fq_all,     // (B, 512)  relu(x@Wq)
    const float* __restrict__ qa,         // (2048, 64) relu(addresses)
    const float* __restrict__ addresses,  // (2048, 64)
    const float* __restrict__ memories,   // (B, 2048, 64)
    float* __restrict__ attn)             // (B, 512)
{
  __shared__ float fk_s[8][64];       // fk[b]; reused for fq[b] after M loop
  __shared__ float v_s[8][128];
  __shared__ float mat_s[64][128];    // matrix; cols 0..63 reused for matrix2
  __shared__ float norm_s[64];
  __shared__ float qa_s[16][65];
  __shared__ float ad_s[16][64];
  __shared__ float den_s[16];
  __shared__ float res_s[16][132];
  __shared__ float mem_s[16][64];
  __shared__ float fkm_s[16][65];
  __shared__ float norm2_s[64];
  __shared__ float memtile[2][16][64];  // TDM double buffer (2 x 4KB)

  const int b    = blockIdx.x;
  const int tid  = threadIdx.x;
  const int lane = tid & 31;
  const int wave = tid >> 5;
  const int mn = lane & 15;
  const int kh = (lane >> 4) << 1;
  const int cm = (lane < 16) ? 0 : 8;

  // kick off TDM load of memories tile 0 (wave 0 only; EXEC ignored by TDM)
  if (wave == 0)
    tdm_load_tile_4k(memories + (size_t)b * 2048 * 64, lds_off(&memtile[0][0][0]));

  // --- load fk[b] (8x64), v[b] (8x128) ---
  ((float*)fk_s)[tid]       = fk_all[(size_t)b * 512 + tid];
  ((float*)fk_s)[tid + 256] = fk_all[(size_t)b * 512 + tid + 256];
#pragma unroll
  for (int j = 0; j < 4; ++j)
    ((float*)v_s)[tid + 256 * j] = v_all[(size_t)b * 1024 + tid + 256 * j];
  __syncthreads();

  if (tid < 64) {  // norm[d] = sum_h fk[h][d]
    float s = 0.0f;
#pragma unroll
    for (int h = 0; h < 8; ++h) s += fk_s[h][tid];
    norm_s[tid] = s;
  }

  // --- matrix = fk^T @ v : wave covers cols [16w,16w+16) ---
#pragma unroll
  for (int db = 0; db < 4; ++db) {
    v8f acc = v8f_zero();
#pragma unroll
    for (int kk = 0; kk < 8; kk += 4) {
      v2f a, bv;
      a.x  = fk_s[kk + kh][db * 16 + mn];
      a.y  = fk_s[kk + kh + 1][db * 16 + mn];
      bv.x = v_s[kk + kh][wave * 16 + mn];
      bv.y = v_s[kk + kh + 1][wave * 16 + mn];
      acc  = wmma_f32_k4(a, bv, acc);
    }
#pragma unroll
    for (int r = 0; r < 8; ++r)
      mat_s[db * 16 + cm + r][wave * 16 + mn] = acc[r];
  }
  __syncthreads();

  // matrix2 accumulators: wave owns tiles (wave) and (wave+8) of 4x4 grid
  v8f m2a = v8f_zero(), m2b = v8f_zero();
  const int db0 = (wave >> 2) << 4;
  const int eb0 = (wave & 3) << 4;
  const int db1 = db0 + 32;
  float n2 = 0.0f;

  for (int mt = 0; mt < 128; ++mt) {
    const int m0 = mt << 4;

    // wave 0: issue TDM for tile mt+1, then ensure tile mt has landed.
    if (wave == 0) {
      if (mt + 1 < 128) {
        tdm_load_tile_4k(memories + ((size_t)b * 2048 + m0 + 16) * 64,
                         lds_off(&memtile[(mt + 1) & 1][0][0]));
        __builtin_amdgcn_s_wait_tensorcnt(1);   // older (mt) load complete
      } else {
        __builtin_amdgcn_s_wait_tensorcnt(0);
      }
    }

    {  // qa / addresses tile (16x64 each)
      int m  = tid >> 4;
      int d4 = (tid & 15) << 2;
      const float* qsrc = qa + (size_t)(m0 + m) * 64 + d4;
      const float* asrc = addresses + (size_t)(m0 + m) * 64 + d4;
#pragma unroll
      for (int j = 0; j < 4; ++j) {
        qa_s[m][d4 + j] = qsrc[j];
        ad_s[m][d4 + j] = asrc[j];
      }
    }
    __syncthreads();  // qa/ad visible; TDM tile mt visible to all waves

    if (tid < 16) {  // den[m] = qa[m] . norm + eps
      float s = 0.0f;
#pragma unroll 8
      for (int d = 0; d < 64; ++d) s += qa_s[tid][d] * norm_s[d];
      den_s[tid] = s + EPSF;
    }

    // num tile: (16x64)@(64x16); batched fragments + dual accumulators
    v8f n0 = v8f_zero(), n1 = v8f_zero();
#pragma unroll
    for (int g = 0; g < 2; ++g) {
      v2f af[8], bf[8];
#pragma unroll
      for (int i = 0; i < 8; ++i) {
        int kb = (g << 5) + (i << 2);
        af[i].x = qa_s[mn][kb + kh];
        af[i].y = qa_s[mn][kb + kh + 1];
        bf[i].x = mat_s[kb + kh][wave * 16 + mn];
        bf[i].y = mat_s[kb + kh + 1][wave * 16 + mn];
      }
#pragma unroll
      for (int i = 0; i < 8; i += 2) {
        n0 = wmma_f32_k4(af[i],     bf[i],     n0);
        n1 = wmma_f32_k4(af[i + 1], bf[i + 1], n1);
      }
    }
    __syncthreads();

    {  // res = num/den; waves 4..7 (e>=64) store sigmoid(res) directly
      int e = wave * 16 + mn;
#pragma unroll
      for (int r = 0; r < 8; ++r) {
        float val = (n0[r] + n1[r]) / den_s[cm + r];
        if (wave >= 4) val = 1.0f / (1.0f + __expf(-val));
        res_s[cm + r][e] = val;
      }
    }
    __syncthreads();

    {  // new_mem blend (memories tile from TDM LDS buffer) + fkm
      int m  = tid >> 4;
      int d4 = (tid & 15) << 2;
      const float* msrc = &memtile[mt & 1][m][d4];
#pragma unroll
      for (int j = 0; j < 4; ++j) {
        float u  = res_s[m][d4 + j];
        float p  = res_s[m][64 + d4 + j];
        float nm = msrc[j] * (1.0f - p) + u * p;
        mem_s[m][d4 + j] = nm;
        fkm_s[m][d4 + j] = fmaxf(nm + ad_s[m][d4 + j], 0.0f);
      }
    }
    __syncthreads();

    // matrix2 += fkm^T @ new_mem : shared B fragments, two WMMA chains
    {
      v2f a0[4], a1[4], bf[4];
#pragma unroll
      for (int i = 0; i < 4; ++i) {
        int kb = i << 2;
        a0[i].x = fkm_s[kb + kh][db0 + mn];
        a0[i].y = fkm_s[kb + kh + 1][db0 + mn];
        a1[i].x = fkm_s[kb + kh][db1 + mn];
        a1[i].y = fkm_s[kb + kh + 1][db1 + mn];
        bf[i].x = mem_s[kb + kh][eb0 + mn];
        bf[i].y = mem_s[kb + kh + 1][eb0 + mn];
      }
#pragma unroll
      for (int i = 0; i < 4; ++i) {
        m2a = wmma_f32_k4(a0[i], bf[i], m2a);
        m2b = wmma_f32_k4(a1[i], bf[i], m2b);
      }
    }
    if (tid < 64) {  // norm2[d] += sum_m fkm[m][d]
      float s = 0.0f;
#pragma unroll
      for (int m = 0; m < 16; ++m) s += fkm_s[m][tid];
      n2 += s;
    }
    __syncthreads();
  }

  // --- matrix2 -> mat_s[0..63][0..63], norm2; fq[b] -> fk_s ---
#pragma unroll
  for (int r = 0; r < 8; ++r) {
    mat_s[db0 + cm + r][eb0 + mn] = m2a[r];
    mat_s[db1 + cm + r][eb0 + mn] = m2b[r];
  }
  if (tid < 64) norm2_s[tid] = n2;
  ((float*)fk_s)[tid]       = fq_all[(size_t)b * 512 + tid];
  ((float*)fk_s)[tid + 256] = fq_all[(size_t)b * 512 + tid + 256];
  __syncthreads();

  // --- attn[b,h,e] = (fq[h].matrix2[:,e]) / (fq[h].norm2 + eps) ---
  const float* fqb = (const float*)fk_s;
#pragma unroll
  for (int t = 0; t < 2; ++t) {
    int idx = tid + t * 256;
    int h = idx >> 6;
    int e = idx & 63;
    float num2 = 0.0f, den2 = 0.0f;
#pragma unroll 8
    for (int d = 0; d < 64; ++d) {
      float f = fqb[(h << 6) + d];
      num2 += f * mat_s[d][e];
      den2 += f * norm2_s[d];
    }
    attn[(size_t)b * 512 + idx] = num2 / (den2 + EPSF);
  }
}

// ---------------------------------------------------------------------------
extern "C" void kernel_launch(void* const* d_in, const int* in_sizes, int n_in,
                              void* d_out, int out_size, void* d_ws, size_t ws_size,
                              hipStream_t stream) {
  (void)in_sizes; (void)n_in; (void)out_size; (void)ws_size;
  const float* x    = (const float*)d_in[0];
  const float* mem  = (const float*)d_in[1];
  const float* addr = (const float*)d_in[2];
  const float* Wk   = (const float*)d_in[3];
  const float* bk   = (const float*)d_in[4];
  const float* Wv   = (const float*)d_in[5];
  const float* bv   = (const float*)d_in[6];
  const float* Wq   = (const float*)d_in[7];
  const float* bq   = (const float*)d_in[8];
  const float* Wm   = (const float*)d_in[9];
  const float* bm   = (const float*)d_in[10];
  float* out = (float*)d_out;

  float* ws   = (float*)d_ws;
  float* fk   = ws;                  // 512*512
  float* vv   = fk + 512 * 512;      // 512*1024
  float* fq   = vv + 512 * 1024;     // 512*512
  float* qa   = fq + 512 * 512;      // 2048*64
  float* attn = qa + 2048 * 64;      // 512*512

  relu_kernel<<<dim3((2048 * 64) / 256), 256, 0, stream>>>(addr, qa, 2048 * 64);
  gemm_bias_act<<<dim3(512 / 64, 512 / 32), 256, 0, stream>>>(x, Wk, bk, fk, 512, 512, 1024, 1);
  gemm_bias_act<<<dim3(1024 / 64, 512 / 32), 256, 0, stream>>>(x, Wv, bv, vv, 512, 1024, 1024, 0);
  gemm_bias_act<<<dim3(512 / 64, 512 / 32), 256, 0, stream>>>(x, Wq, bq, fq, 512, 512, 1024, 1);
  fused_memory_kernel<<<dim3(512), 256, 0, stream>>>(fk, vv, fq, qa, addr, mem, attn);
  gemm_bias_act<<<dim3(1024 / 64, 512 / 32), 256, 0, stream>>>(attn, Wm, bm, out, 512, 1024, 512, 0);
}